// LSTMPointerNet_944892805495
// MI455X (gfx1250) — compile-verified
//
#include <hip/hip_runtime.h>
#include <math.h>

typedef float v2f __attribute__((ext_vector_type(2)));
typedef float v8f __attribute__((ext_vector_type(8)));

#define HID 512
#define NEGV -1e9f

__device__ __forceinline__ float sigmoidf_(float x) {
    return 1.0f / (1.0f + expf(-x));
}

// ---------------------------------------------------------------------------
// K1: query = one LSTM step with h0=c0=0.
// gates = x @ w_ih.T + b_ih + b_hh ; f-gate irrelevant (f*c0 = 0).
// h = sigmoid(o) * tanh( sigmoid(i) * tanh(g) )
// ---------------------------------------------------------------------------
__global__ __launch_bounds__(256)
void query_kernel(const float* __restrict__ ptr_in,   // [B,512]
                  const float* __restrict__ w_ih,     // [2048,512]
                  const float* __restrict__ b_ih,     // [2048]
                  const float* __restrict__ b_hh,     // [2048]
                  float* __restrict__ query)          // [B,512]
{
    const int b = blockIdx.x;
    __shared__ float x[HID];
    for (int i = threadIdx.x; i < HID; i += 256) x[i] = ptr_in[b * HID + i];
    __syncthreads();

    for (int k = threadIdx.x; k < HID; k += 256) {
        const float* wi = w_ih + (size_t)(k) * HID;          // i-gate row
        const float* wg = w_ih + (size_t)(1024 + k) * HID;   // g-gate row
        const float* wo = w_ih + (size_t)(1536 + k) * HID;   // o-gate row
        float gi = b_ih[k]        + b_hh[k];
        float gg = b_ih[1024 + k] + b_hh[1024 + k];
        float go = b_ih[1536 + k] + b_hh[1536 + k];
        for (int d = 0; d < HID; ++d) {
            float xv = x[d];
            gi = fmaf(xv, wi[d], gi);
            gg = fmaf(xv, wg[d], gg);
            go = fmaf(xv, wo[d], go);
        }
        float c = sigmoidf_(gi) * tanhf(gg);
        query[b * HID + k] = sigmoidf_(go) * tanhf(c);
    }
}

// ---------------------------------------------------------------------------
// Batched GEMV: out[b,:] = (addv ? addv[b,:] : 0) + rows[b,:] @ W
// ---------------------------------------------------------------------------
__global__ __launch_bounds__(256)
void gemv_kernel(const float* __restrict__ rows,   // [B,512]
                 const float* __restrict__ W,      // [512,512]
                 const float* __restrict__ addv,   // [B,512] or nullptr
                 float* __restrict__ out)          // [B,512]
{
    const int b = blockIdx.x;
    __shared__ float q[HID];
    for (int i = threadIdx.x; i < HID; i += 256) q[i] = rows[b * HID + i];
    __syncthreads();

    for (int h = threadIdx.x; h < HID; h += 256) {
        float s = addv ? addv[b * HID + h] : 0.0f;
        for (int d = 0; d < HID; ++d)
            s = fmaf(q[d], W[d * HID + h], s);
        out[b * HID + h] = s;
    }
}

// ---------------------------------------------------------------------------
// Fused feature+score kernel (WMMA workhorse), 32 rows per block.
// feat = rows @ W (32x512, K=512) via v_wmma_f32_16x16x4_f32, then
// score[row] = sum_n tanh(feat[row,n] + bias[b,n]) * v[n].
//
// Block = 256 threads = 8 waves. Wave w owns cols [64w, 64w+64) as 4
// N-tiles and BOTH 16-row M-tiles, so every B fragment (2 dwords/lane from
// L2) feeds two WMMAs -> 1 global load per wmma instead of 2, and B L2
// traffic is halved vs a 16-row block. A tile lives in LDS with a 516-float
// pitch so A-fragment fetches are bank-conflict-free ds_load_2addr_b64.
// ---------------------------------------------------------------------------
__global__ __launch_bounds__(256)
void fused_feat_score_kernel(const float* __restrict__ mem,   // [B,R,512]
                             const float* __restrict__ W,     // [512,512]
                             const float* __restrict__ biasv, // [B,512]
                             const float* __restrict__ vvec,  // [512]
                             const int*   __restrict__ lens,  // [B] or nullptr
                             float* __restrict__ out,         // [B,R]
                             int R, int applyMask)
{
    const int b  = blockIdx.y;
    const int s0 = blockIdx.x * 32;

    __shared__ float As[32][516];   // 32x512 A tile, padded pitch
    __shared__ float red[8][32];    // per-wave partial scores

    const int tid = threadIdx.x;
    const float* A = mem + ((size_t)b * R + s0) * HID;
    for (int idx = tid; idx < 32 * HID; idx += 256) {
        int r = idx >> 9, c = idx & 511;
        As[r][c] = A[r * HID + c];
    }
    __syncthreads();

    const int w    = tid >> 5;      // wave id 0..7
    const int l    = tid & 31;      // lane
    const int m    = l & 15;        // M-in-tile for A frag / N-in-tile for B,C
    const int half = l >> 4;        // selects K pair (A,B) / M half (C)

    v8f acc0[4] = {};               // M-tile 0 (rows 0..15)
    v8f acc1[4] = {};               // M-tile 1 (rows 16..31)

    for (int k4 = 0; k4 < HID; k4 += 4) {
        const int krow = k4 + 2 * half;
        v2f a0, a1;
        a0.x = As[m][krow];
        a0.y = As[m][krow + 1];
        a1.x = As[16 + m][krow];
        a1.y = As[16 + m][krow + 1];
#pragma unroll
        for (int j = 0; j < 4; ++j) {
            const int n = w * 64 + j * 16 + m;
            v2f bb;
            bb.x = W[(size_t)krow * HID + n];
            bb.y = W[(size_t)(krow + 1) * HID + n];
            acc0[j] = __builtin_amdgcn_wmma_f32_16x16x4_f32(
                false, a0, false, bb, (short)0, acc0[j], false, false);
            acc1[j] = __builtin_amdgcn_wmma_f32_16x16x4_f32(
                false, a1, false, bb, (short)0, acc1[j], false, false);
        }
    }

    // epilogue: partial scores. C layout: lane l holds N = l&15,
    // rows j + 8*(l>>4) in vgpr j -> p[mt][v] is row mt*16 + 8*half + v.
    float p0[8], p1[8];
#pragma unroll
    for (int v = 0; v < 8; ++v) { p0[v] = 0.0f; p1[v] = 0.0f; }
#pragma unroll
    for (int j = 0; j < 4; ++j) {
        const int n = w * 64 + j * 16 + m;
        const float bn = biasv[b * HID + n];
        const float vn = vvec[n];
#pragma unroll
        for (int v = 0; v < 8; ++v) {
            p0[v] += tanhf(acc0[j][v] + bn) * vn;
            p1[v] += tanhf(acc1[j][v] + bn) * vn;
        }
    }
    // reduce over the 16 lanes of each half (different n within tile)
#pragma unroll
    for (int off = 1; off < 16; off <<= 1) {
#pragma unroll
        for (int v = 0; v < 8; ++v) {
            p0[v] += __shfl_xor(p0[v], off, 32);
            p1[v] += __shfl_xor(p1[v], off, 32);
        }
    }
    if (m == 0) {  // lanes 0 (rows 0-7 per tile) and 16 (rows 8-15 per tile)
#pragma unroll
        for (int v = 0; v < 8; ++v) {
            red[w][half * 8 + v]      = p0[v];
            red[w][16 + half * 8 + v] = p1[v];
        }
    }
    __syncthreads();

    if (tid < 32) {
        float s = 0.0f;
#pragma unroll
        for (int ww = 0; ww < 8; ++ww) s += red[ww][tid];
        const int row = s0 + tid;
        if (applyMask && row >= lens[b]) s = NEGV;
        out[(size_t)b * R + row] = s;
    }
}

// ---------------------------------------------------------------------------
// K4: masked softmax over entity scores + t[b,d] = attn[b,:] @ entity_mem[b]
// (ctx is then t @ side_wm, so side_feat is never materialized.)
// ---------------------------------------------------------------------------
__global__ __launch_bounds__(256)
void entity_softmax_t_kernel(const float* __restrict__ escore,     // [B,256]
                             const int*   __restrict__ entity_nums,// [B]
                             const float* __restrict__ entity_mem, // [B,256,512]
                             float* __restrict__ t)                // [B,512]
{
    const int b = blockIdx.x;
    const int tid = threadIdx.x;
    __shared__ float sc[256];
    __shared__ float redu[256];

    const int n = entity_nums[b];
    float v = escore[b * 256 + tid];
    if (tid >= n) v = NEGV;

    redu[tid] = v;
    __syncthreads();
    for (int off = 128; off > 0; off >>= 1) {
        if (tid < off) redu[tid] = fmaxf(redu[tid], redu[tid + off]);
        __syncthreads();
    }
    const float mx = redu[0];
    __syncthreads();

    const float e = expf(v - mx);
    redu[tid] = e;
    __syncthreads();
    for (int off = 128; off > 0; off >>= 1) {
        if (tid < off) redu[tid] += redu[tid + off];
        __syncthreads();
    }
    const float sum = redu[0];
    __syncthreads();

    sc[tid] = e / sum;
    __syncthreads();

    const float* em = entity_mem + (size_t)b * 256 * HID;
    for (int d = tid; d < HID; d += 256) {
        float acc = 0.0f;
        for (int e2 = 0; e2 < 256; ++e2)
            acc = fmaf(sc[e2], em[(size_t)e2 * HID + d], acc);
        t[b * HID + d] = acc;
    }
}

// ---------------------------------------------------------------------------
extern "C" void kernel_launch(void* const* d_in, const int* in_sizes, int n_in,
                              void* d_out, int out_size, void* d_ws, size_t ws_size,
                              hipStream_t stream) {
    const float* sent_mem    = (const float*)d_in[0];   // [64,1024,512]
    const float* entity_mem  = (const float*)d_in[1];   // [64,256,512]
    const float* ptr_in      = (const float*)d_in[2];   // [64,1,512]
    const int*   sent_nums   = (const int*)  d_in[3];   // [64]
    const int*   entity_nums = (const int*)  d_in[4];   // [64]
    const float* lstm_w_ih   = (const float*)d_in[5];   // [2048,512]
    // d_in[6] = lstm_w_hh: unused (h0 = 0)
    const float* lstm_b_ih   = (const float*)d_in[7];   // [2048]
    const float* lstm_b_hh   = (const float*)d_in[8];   // [2048]
    const float* attn_wm     = (const float*)d_in[9];   // [512,512]
    const float* attn_wq     = (const float*)d_in[10];  // [512,512]
    const float* attn_v      = (const float*)d_in[11];  // [512]
    const float* side_wm     = (const float*)d_in[12];  // [512,512]
    const float* side_wq     = (const float*)d_in[13];  // [512,512]
    const float* side_v      = (const float*)d_in[14];  // [512]
    const float* attn_ws     = (const float*)d_in[15];  // [512,512]

    float* out = (float*)d_out;      // [64,1,1024] flat
    float* ws  = (float*)d_ws;

    float* query  = ws;              // 64*512
    float* qwq    = ws + 32768;      // 64*512  query @ side_wq
    float* qattn  = ws + 65536;      // 64*512  query @ attn_wq
    float* escore = ws + 98304;      // 64*256  raw entity scores
    float* tvec   = ws + 114688;     // 64*512  attn @ entity_mem
    float* ctx    = ws + 147456;     // 64*512
    float* biasv  = ws + 180224;     // 64*512  qattn + ctx @ attn_ws

    // 1) LSTM single step -> query
    query_kernel<<<64, 256, 0, stream>>>(ptr_in, lstm_w_ih, lstm_b_ih, lstm_b_hh, query);

    // 2) query projections
    gemv_kernel<<<64, 256, 0, stream>>>(query, side_wq, nullptr, qwq);
    gemv_kernel<<<64, 256, 0, stream>>>(query, attn_wq, nullptr, qattn);

    // 3) entity attention scores: tanh(entity_mem@side_wm + qwq) @ side_v (WMMA)
    fused_feat_score_kernel<<<dim3(8, 64), 256, 0, stream>>>(
        entity_mem, side_wm, qwq, side_v, nullptr, escore, 256, 0);

    // 4) masked softmax + t = attn @ entity_mem
    entity_softmax_t_kernel<<<64, 256, 0, stream>>>(escore, entity_nums, entity_mem, tvec);

    // 5) ctx = t @ side_wm ; bias = qattn + ctx @ attn_ws
    gemv_kernel<<<64, 256, 0, stream>>>(tvec, side_wm, nullptr, ctx);
    gemv_kernel<<<64, 256, 0, stream>>>(ctx, attn_ws, qattn, biasv);

    // 6) pointer scores over sentences (dominant WMMA kernel), masked -> out
    fused_feat_score_kernel<<<dim3(32, 64), 256, 0, stream>>>(
        sent_mem, attn_wm, biasv, attn_v, sent_nums, out, 1024, 1);
}